// DownSample_14937896256227
// MI455X (gfx1250) — compile-verified
//
#include <hip/hip_runtime.h>

typedef __attribute__((ext_vector_type(2))) float v2f;
typedef __attribute__((ext_vector_type(8))) float v8f;

#define N_IN   2000000
#define N_OUT  250000
#define KOCT   8
#define CIN    32
#define COUT   64
#define NTILES (N_OUT / 16)          // 15625, exact
#define WPB    8                     // waves per block (256 threads, wave32)
#define BN_EPS 1e-5f

// ---------------------------------------------------------------------------
// Kernel 0: zero the global BN accumulators (sum[64], sumsq[64]) in d_ws.
// ---------------------------------------------------------------------------
__global__ void zero_acc_kernel(float* __restrict__ acc) {
    if (threadIdx.x < 128) acc[threadIdx.x] = 0.0f;
}

// ---------------------------------------------------------------------------
// Kernel 1: gather + GEMM via V_WMMA_F32_16X16X4_F32, fused BN partial stats.
// Each wave computes a 16x64 output tile: C[16x64] = A[16x256] * W[256x64].
// A-fragment (16x4 f32, 2 VGPRs): lanes 0-15 hold M=lane, K = k,k+1;
//                                 lanes 16-31 hold M=lane-16, K = k+2,k+3.
// B-fragment (4x16 f32, 2 VGPRs): lanes 0-15 hold N=lane, K = k,k+1;
//                                 lanes 16-31 hold N=lane-16, K = k+2,k+3.
// Weight is staged in LDS pre-swizzled into B-fragment order => ds_load_b64.
// ---------------------------------------------------------------------------
__global__ __launch_bounds__(256)
void gemm_bnstat_kernel(const float* __restrict__ x,
                        const int*   __restrict__ neigh,
                        const float* __restrict__ weight,
                        float*       __restrict__ out,
                        float*       __restrict__ acc_g) {
    // [kstep(64)][jtile(4)][lane(32)][2 floats] = 16384 floats = 64 KB
    __shared__ float wlds[64 * 4 * 32 * 2];
    __shared__ float lds_acc[128];   // sum[64], sumsq[64]

    const int tid = threadIdx.x;

    for (int i = tid; i < 128; i += 256) lds_acc[i] = 0.0f;

    // Stage weight into LDS in B-fragment layout.
    for (int i = tid; i < 64 * 4 * 32 * 2; i += 256) {
        const int kstep = i >> 8;          // 0..63
        const int rem   = i & 255;
        const int j     = rem >> 6;        // n-tile 0..3
        const int rem2  = rem & 63;
        const int ln    = rem2 >> 1;       // lane 0..31
        const int r     = rem2 & 1;        // vgpr 0/1
        const int kk    = kstep >> 3;                          // octant 0..7
        const int c     = ((kstep & 7) << 2) + ((ln >= 16) ? 2 : 0) + r;
        const int d     = (j << 4) + (ln & 15);
        wlds[i] = weight[(kk * CIN + c) * COUT + d];
    }
    __syncthreads();

    const int  wave  = tid >> 5;
    const int  lane  = tid & 31;
    const int  tile  = blockIdx.x * WPB + wave;
    const bool valid = (tile < NTILES);

    if (valid) {
        const int m   = lane & 15;
        const int row = tile * 16 + m;

        // Byte offsets of the 8 gathered x-rows for this lane's output row.
        unsigned off[KOCT];
#pragma unroll
        for (int kk = 0; kk < KOCT; ++kk)
            off[kk] = (unsigned)neigh[row * KOCT + kk] * (unsigned)(CIN * 4);

        const unsigned hi = (lane >= 16) ? 8u : 0u;   // K+2,K+3 for upper half
        const char* xb = (const char*)x;

        v8f acc[4] = {v8f{}, v8f{}, v8f{}, v8f{}};

#pragma unroll
        for (int kk = 0; kk < KOCT; ++kk) {
            const char* rowp = xb + (size_t)off[kk] + hi;
#pragma unroll
            for (int cs = 0; cs < CIN; cs += 4) {
                const v2f a = *(const v2f*)(rowp + cs * 4);
                const int kstep = (kk << 3) + (cs >> 2);
#pragma unroll
                for (int j = 0; j < 4; ++j) {
                    const v2f b =
                        *(const v2f*)&wlds[(((kstep << 2) + j) * 32 + lane) * 2];
                    acc[j] = __builtin_amdgcn_wmma_f32_16x16x4_f32(
                        /*neg_a=*/false, a, /*neg_b=*/false, b,
                        /*c_mod=*/(short)0, acc[j],
                        /*reuse_a=*/false, /*reuse_b=*/false);
                }
            }
        }

        // Store C tile + accumulate per-column partial sum / sum-of-squares.
        // C/D layout: VGPR v: lanes 0-15 -> (M=v, N=lane); lanes 16-31 -> (M=8+v).
        float* op = out + (size_t)tile * 16 * COUT;
        const int rbase = (lane >= 16) ? 8 : 0;
        const int nlo   = lane & 15;
#pragma unroll
        for (int j = 0; j < 4; ++j) {
            float s = 0.0f, q = 0.0f;
#pragma unroll
            for (int v = 0; v < 8; ++v) {
                const float t = acc[j][v];
                op[(rbase + v) * COUT + (j << 4) + nlo] = t;
                s += t;
                q += t * t;
            }
            atomicAdd(&lds_acc[(j << 4) + nlo], s);        // ds_add_f32
            atomicAdd(&lds_acc[64 + (j << 4) + nlo], q);   // ds_add_f32
        }
    }
    __syncthreads();

    // One global float atomic per column per block.
    if (tid < 128) atomicAdd(&acc_g[tid], lds_acc[tid]);
}

// ---------------------------------------------------------------------------
// Kernel 2: finalize BN parameters: scale = gamma*rsqrt(var+eps),
//           shift = beta - mean*scale.
// ---------------------------------------------------------------------------
__global__ void bn_finalize_kernel(const float* __restrict__ acc,
                                   const float* __restrict__ gamma,
                                   const float* __restrict__ beta,
                                   float* __restrict__ params) {
    const int t = threadIdx.x;
    if (t < COUT) {
        const float inv_n = 1.0f / (float)N_OUT;
        const float mean  = acc[t] * inv_n;
        const float var   = acc[64 + t] * inv_n - mean * mean;
        const float sc    = gamma[t] * rsqrtf(var + BN_EPS);
        params[t]      = sc;
        params[64 + t] = beta[t] - mean * sc;
    }
}

// ---------------------------------------------------------------------------
// Kernel 3: apply BN in place on the GEMM output (float4 grid-stride).
// Grid stride is a multiple of 16 float4s, so each thread's column group is
// fixed: cols 4*(tid%16) .. 4*(tid%16)+3.
// ---------------------------------------------------------------------------
__global__ __launch_bounds__(256)
void bn_apply_kernel(float* __restrict__ out,
                     const float* __restrict__ params, int n4) {
    const float4* p = (const float4*)params;
    const float4 s = p[threadIdx.x & 15];
    const float4 b = p[16 + (threadIdx.x & 15)];
    float4* o = (float4*)out;
    const int stride = gridDim.x * blockDim.x;
    for (int i = blockIdx.x * blockDim.x + threadIdx.x; i < n4; i += stride) {
        float4 v = o[i];
        v.x = v.x * s.x + b.x;
        v.y = v.y * s.y + b.y;
        v.z = v.z * s.z + b.z;
        v.w = v.w * s.w + b.w;
        o[i] = v;
    }
}

// ---------------------------------------------------------------------------
extern "C" void kernel_launch(void* const* d_in, const int* in_sizes, int n_in,
                              void* d_out, int out_size, void* d_ws, size_t ws_size,
                              hipStream_t stream) {
    const float* x      = (const float*)d_in[0];   // [2e6, 32]
    const int*   neigh  = (const int*)d_in[1];     // [250k, 8]
    const float* weight = (const float*)d_in[2];   // [8, 32, 64]
    const float* gamma  = (const float*)d_in[3];   // [64]
    const float* beta   = (const float*)d_in[4];   // [64]
    float* out = (float*)d_out;                    // [250k, 64]

    float* acc    = (float*)d_ws;                  // 128 floats
    float* params = acc + 128;                     // 128 floats

    zero_acc_kernel<<<1, 128, 0, stream>>>(acc);

    const int blocks = (NTILES + WPB - 1) / WPB;   // 1954
    gemm_bnstat_kernel<<<blocks, 256, 0, stream>>>(x, neigh, weight, out, acc);

    bn_finalize_kernel<<<1, 64, 0, stream>>>(acc, gamma, beta, params);

    const int n4 = N_OUT * COUT / 4;               // 4,000,000
    bn_apply_kernel<<<4096, 256, 0, stream>>>(out, params, n4);
}